// NormalLoss_10934986736189
// MI455X (gfx1250) — compile-verified
//
#include <hip/hip_runtime.h>
#include <hip/hip_bf16.h>

// ---------------------------------------------------------------------------
// NormalLoss for MI455X (gfx1250, wave32, WMMA)
//
// Phase 1: per-pred nearest gt (LDS-tiled argmin); emit q4[j]={qx,qy,qz,pq}.
// Phase 2: stream 256MB adjacency once (HBM-bound, ~11us floor @23.3TB/s).
//          One V_WMMA_F32_16X16X4_F32 per 16x16 tile computes
//          d = preds_i . q_j - pq_j directly (pq folded into the K=3 pad
//          slot: A[.,3]=1, B[3,.]=-pq). Adjacency loads are non-temporal
//          (256MB > 192MB L2); prefetch runs 8 tiles ahead, speculative.
// ---------------------------------------------------------------------------

typedef __attribute__((ext_vector_type(2)))  float v2f;
typedef __attribute__((ext_vector_type(8)))  float v8f;

#define N_PRED   8192
#define N_GT     8192
#define TILE_GTS 2048   // 2048 gts * 4 floats (x,y,z,|g|^2) = 32KB LDS
#define J_CHUNK  32     // column tiles per wave

// ---------------------------------------------------------------------------
// Phase 1: nearest-gt argmin per pred; build q4 = {qx,qy,qz, p.q}. Zero out.
// ---------------------------------------------------------------------------
__global__ void __launch_bounds__(256)
nn_phase1(const float* __restrict__ preds,
          const float* __restrict__ gts,
          const float* __restrict__ gtsn,
          float4* __restrict__ q4,
          float* __restrict__ out)
{
    __shared__ float sh[TILE_GTS * 4];

    const int j = blockIdx.x * blockDim.x + threadIdx.x;   // pred index
    if (j == 0) out[0] = 0.0f;                             // runs before phase 2

    const float px = preds[j * 3 + 0];
    const float py = preds[j * 3 + 1];
    const float pz = preds[j * 3 + 2];

    float best = 3.402823466e38f;
    int   bi   = 0;

    for (int t0 = 0; t0 < N_GT; t0 += TILE_GTS) {
        __syncthreads();
        for (int t = threadIdx.x; t < TILE_GTS; t += blockDim.x) {
            const float gx = gts[(t0 + t) * 3 + 0];
            const float gy = gts[(t0 + t) * 3 + 1];
            const float gz = gts[(t0 + t) * 3 + 2];
            sh[t * 4 + 0] = gx;
            sh[t * 4 + 1] = gy;
            sh[t * 4 + 2] = gz;
            sh[t * 4 + 3] = gx * gx + gy * gy + gz * gz;
        }
        __syncthreads();
        // argmin_i (|g_i|^2 - 2 g_i.p_j)   (|p_j|^2 constant in i, dropped)
        for (int t = 0; t < TILE_GTS; ++t) {
            const float m = sh[t * 4 + 3] -
                2.0f * (sh[t * 4 + 0] * px + sh[t * 4 + 1] * py + sh[t * 4 + 2] * pz);
            if (m < best) { best = m; bi = t0 + t; }   // strict < keeps first index
        }
    }

    const float qx = gtsn[bi * 3 + 0];
    const float qy = gtsn[bi * 3 + 1];
    const float qz = gtsn[bi * 3 + 2];
    float4 v;
    v.x = qx; v.y = qy; v.z = qz;
    v.w = px * qx + py * qy + pz * qz;   // pq[j]
    q4[j] = v;
}

// ---------------------------------------------------------------------------
// Phase 2.
// f32 16x4 A layout (ISA 7.12.2): v0 = K=0 (lanes 0-15) / K=2 (lanes 16-31),
// v1 = K=1 / K=3. B (4x16) mirrored. C: reg r, lanes 0-15 -> (M=r, N=lane),
// lanes 16-31 -> (M=r+8, N=lane-16) — matches the adjacency tile load order.
// ---------------------------------------------------------------------------
__global__ void __launch_bounds__(256)
loss_phase2(const float* __restrict__ preds,
            const float* __restrict__ adj,
            const float4* __restrict__ q4,
            float* __restrict__ out)
{
    const int lane = threadIdx.x & 31;
    const int wave = (blockIdx.x * blockDim.x + threadIdx.x) >> 5;
    const int chunksPerRow = (N_PRED / 16) / J_CHUNK;      // 16
    const int I  = wave / chunksPerRow;                    // row tile 0..511
    const int j0 = (wave % chunksPerRow) * J_CHUNK;        // first column tile
    const int hi  = lane >> 4;                             // 0: lanes 0-15, 1: 16-31
    const int l15 = lane & 15;

    // --- A operand: preds rows I*16 .. I*16+15; K=3 slot carries 1.0 so the
    //     B-side K=3 row (-pq_j) is added by the WMMA itself.
    const int irow = I * 16 + l15;
    v2f amat;
    amat.x = hi ? preds[irow * 3 + 2] : preds[irow * 3 + 0];
    amat.y = hi ? 1.0f                : preds[irow * 3 + 1];

    // Adjacency row pointers matching the C layout (reg r -> row r + 8*hi)
    const float* rowp[8];
#pragma unroll
    for (int r = 0; r < 8; ++r)
        rowp[r] = adj + (size_t)(I * 16 + r + 8 * hi) * (size_t)N_PRED;

    float acc0 = 0.0f, acc1 = 0.0f;

    for (int jt = j0; jt < j0 + J_CHUNK; ++jt) {
        const int jcol = jt * 16 + l15;

        // Speculative prefetch 8 tiles (512B) ahead on the adjacency stream;
        // unconditional: OOB speculative prefetches are dropped (ISA 7.3).
#pragma unroll
        for (int r = 0; r < 8; ++r)
            __builtin_prefetch(rowp[r] + jcol + 8 * 16, 0, 0);

        // --- B operand: one b128 pulls {qx,qy,qz,pq}; K=3 row = -pq_j ---
        const float4 qv = q4[jcol];
        v2f bmat;
        bmat.x = hi ? qv.z  : qv.x;
        bmat.y = hi ? -qv.w : qv.y;

        v8f cmat = {};
        cmat = __builtin_amdgcn_wmma_f32_16x16x4_f32(
            /*neg_a=*/false, amat, /*neg_b=*/false, bmat,
            /*c_mod=*/(short)0, cmat, /*reuse_a=*/false, /*reuse_b=*/false);
        // cmat[r] = preds_i . q_j - pq_j   for (i = I*16 + r + 8*hi, j = jcol)

        // --- elementwise: acc += (A[i,j] * d)^2 ; NT loads keep L2 clean ---
#pragma unroll
        for (int r = 0; r < 8; r += 2) {
            const float av0 = __builtin_nontemporal_load(rowp[r]     + jcol);
            const float av1 = __builtin_nontemporal_load(rowp[r + 1] + jcol);
            const float t0 = av0 * cmat[r];
            const float t1 = av1 * cmat[r + 1];
            acc0 = fmaf(t0, t0, acc0);
            acc1 = fmaf(t1, t1, acc1);
        }
    }

    float acc = acc0 + acc1;
    // wave32 reduction then one atomic per wave
#pragma unroll
    for (int off = 16; off >= 1; off >>= 1)
        acc += __shfl_xor(acc, off, 32);
    if (lane == 0)
        atomicAdd(out, acc);
}

// ---------------------------------------------------------------------------
// Launch
// ---------------------------------------------------------------------------
extern "C" void kernel_launch(void* const* d_in, const int* in_sizes, int n_in,
                              void* d_out, int out_size, void* d_ws, size_t ws_size,
                              hipStream_t stream) {
    (void)in_sizes; (void)n_in; (void)out_size; (void)ws_size;

    const float* preds = (const float*)d_in[0];   // (8192, 3)
    const float* gts   = (const float*)d_in[1];   // (8192, 3)
    const float* gtsn  = (const float*)d_in[2];   // (8192, 3)
    const float* adj   = (const float*)d_in[3];   // (8192, 8192)
    float* out = (float*)d_out;

    float4* q4 = (float4*)d_ws;                   // 8192 * 16B = 128KB scratch

    // Phase 1: 8192 threads, one per pred
    hipLaunchKernelGGL(nn_phase1, dim3(N_PRED / 256), dim3(256), 0, stream,
                       preds, gts, gtsn, q4, out);

    // Phase 2: 512 row tiles * 16 chunks = 8192 waves = 1024 blocks of 256
    const int waves  = (N_PRED / 16) * ((N_PRED / 16) / J_CHUNK);
    const int blocks = waves * 32 / 256;
    hipLaunchKernelGGL(loss_phase2, dim3(blocks), dim3(256), 0, stream,
                       preds, adj, q4, out);
}